// PaiNNEncoder_48206712930392
// MI455X (gfx1250) — compile-verified
//
#include <hip/hip_runtime.h>
#include <math.h>

#define DEV __device__ __forceinline__

constexpr int Nn = 20000;   // nodes
constexpr int Ee = 320000;  // edges
constexpr int Dd = 128;     // feature dim
constexpr int RB = 64;      // rbf count
constexpr int Ll = 3;       // layers

typedef __attribute__((ext_vector_type(2))) float v2f;
typedef __attribute__((ext_vector_type(8))) float v8f;

// ---------------- WMMA helpers (fp32 16x16x4, CDNA5) ----------------
DEV v8f wmma4(v2f a, v2f b, v8f c) {
  return __builtin_amdgcn_wmma_f32_16x16x4_f32(false, a, false, b, (short)0, c,
                                               false, false);
}

// A fragment: row-major [16 x K], K-chunk k0 (width 4).
// ISA layout: VGPR0 = {K=k0 lanes0-15, K=k0+2 lanes16-31}, VGPR1 = +1.
DEV v2f frag_a(const float* A, int lda, int k0, int lane) {
  int m = lane & 15;
  int k = k0 + ((lane >> 4) << 1);
  v2f a;
  a.x = A[(size_t)m * lda + k];
  a.y = A[(size_t)m * lda + k + 1];
  return a;
}

// B fragment: row-major [K x N], K-chunk k0, col tile n0 (symmetric to A).
DEV v2f frag_b(const float* B, int ldb, int k0, int n0, int lane) {
  int n = n0 + (lane & 15);
  int k = k0 + ((lane >> 4) << 1);
  v2f b;
  b.x = B[(size_t)k * ldb + n];
  b.y = B[(size_t)(k + 1) * ldb + n];
  return b;
}

// C/D layout: VGPR r -> row (r + 8*(lane>=16)), col (lane&15)
DEV int crow(int r, int lane) { return r + ((lane >> 4) << 3); }

DEV float silu(float v) { return v / (1.0f + __expf(-v)); }

// Native CDNA5 fp32 scatter-add: GLOBAL_ATOMIC_ADD_F32, non-returning,
// device scope so cross-WGP accumulation resolves at L2 (RMW-at-scope rule).
DEV void atomicAddF32(float* p, float v) {
  asm volatile("global_atomic_add_f32 %0, %1, off scope:SCOPE_DEV"
               :
               : "v"(p), "v"(v)
               : "memory");
}

// ---------------- K1: edge geometry + RBF + filter GEMM ----------------
// Wij[e][0:384] = clip(nan0((phi @ filt_w + b)) * fcut, -5, 5); dir[e][0:3]
__global__ __launch_bounds__(128) void k_edge_filters(
    const float* __restrict__ pos, const int* __restrict__ eidx,
    const float* __restrict__ fw, const float* __restrict__ fb,
    float* __restrict__ Wij, float* __restrict__ dir) {
  __shared__ float phiS[4][16][RB + 4];
  __shared__ float dS[4][16];
  __shared__ float fcS[4][16];
  int wave = threadIdx.x >> 5, lane = threadIdx.x & 31;
  int tile = blockIdx.x * 4 + wave;  // 20000 tiles exactly
  int eb = tile * 16;
  if (lane < 16) {
    int e = eb + lane;
    int i = eidx[e], j = eidx[Ee + e];
    float rx = pos[j * 3 + 0] - pos[i * 3 + 0];
    float ry = pos[j * 3 + 1] - pos[i * 3 + 1];
    float rz = pos[j * 3 + 2] - pos[i * 3 + 2];
    float d = fmaxf(sqrtf(rx * rx + ry * ry + rz * rz), 1e-8f);
    float inv = 1.0f / d;
    dir[(size_t)e * 3 + 0] = rx * inv;
    dir[(size_t)e * 3 + 1] = ry * inv;
    dir[(size_t)e * 3 + 2] = rz * inv;
    float xc = fminf(fmaxf(d * 0.2f, 0.0f), 1.0f);
    fcS[wave][lane] =
        (xc < 1.0f) ? 0.5f * (__cosf(3.14159265358979323f * xc) + 1.0f) : 0.0f;
    dS[wave][lane] = d;
  }
  __syncthreads();
  const float coeff = -0.5f * (63.0f / 5.0f) * (63.0f / 5.0f);
  for (int idx = lane; idx < 16 * RB; idx += 32) {
    int m = idx >> 6, r = idx & 63;
    float t = dS[wave][m] - (5.0f / 63.0f) * (float)r;
    phiS[wave][m][r] = __expf(coeff * t * t);
  }
  __syncthreads();
  for (int nt = 0; nt < 24; ++nt) {
    int n0 = nt * 16;
    float bn = fb[n0 + (lane & 15)];
    v8f c;
    for (int r = 0; r < 8; ++r) c[r] = bn;
#pragma unroll 4
    for (int ks = 0; ks < 16; ++ks) {
      v2f a = frag_a(&phiS[wave][0][0], RB + 4, ks * 4, lane);
      v2f b = frag_b(fw, 384, ks * 4, n0, lane);
      c = wmma4(a, b, c);
    }
    for (int r = 0; r < 8; ++r) {
      int m = crow(r, lane), n = lane & 15;
      float v = c[r] * fcS[wave][m];
      v = (v != v) ? 0.0f : fminf(fmaxf(v, -5.0f), 5.0f);
      Wij[(size_t)(eb + m) * 384 + n0 + n] = v;
    }
  }
}

// ---------------- K2: q = silu(z @ in_w + in_b) ----------------
__global__ __launch_bounds__(128) void k_embed(const float* __restrict__ z,
                                               const float* __restrict__ w,
                                               const float* __restrict__ b,
                                               float* __restrict__ q) {
  int wave = threadIdx.x >> 5, lane = threadIdx.x & 31;
  int tile = blockIdx.x * 4 + wave;
  if (tile >= Nn / 16) return;
  int nb = tile * 16;
  const float* A = z + (size_t)nb * 64;
  for (int nt = 0; nt < 8; ++nt) {
    int n0 = nt * 16;
    float bn = b[n0 + (lane & 15)];
    v8f c;
    for (int r = 0; r < 8; ++r) c[r] = bn;
#pragma unroll 4
    for (int ks = 0; ks < 16; ++ks) {
      v2f a = frag_a(A, 64, ks * 4, lane);
      v2f bb = frag_b(w, 128, ks * 4, n0, lane);
      c = wmma4(a, bb, c);
    }
    for (int r = 0; r < 8; ++r) {
      int m = crow(r, lane), n = lane & 15;
      q[(size_t)(nb + m) * 128 + n0 + n] = silu(c[r]);
    }
  }
}

// ---------------- K3: x = silu(q@W1+b1)@W2+b2 ----------------
__global__ __launch_bounds__(128) void k_inter(
    const float* __restrict__ q, const float* __restrict__ W1,
    const float* __restrict__ b1, const float* __restrict__ W2,
    const float* __restrict__ b2, float* __restrict__ x) {
  constexpr int LDH = 132;
  __shared__ float hS[4][16 * LDH];
  int wave = threadIdx.x >> 5, lane = threadIdx.x & 31;
  int tile = blockIdx.x * 4 + wave;
  bool active = tile < Nn / 16;
  int nb = tile * 16;
  const float* A = q + (size_t)nb * 128;
  if (active) {
    for (int nt = 0; nt < 8; ++nt) {
      int n0 = nt * 16;
      float bn = b1[n0 + (lane & 15)];
      v8f c;
      for (int r = 0; r < 8; ++r) c[r] = bn;
#pragma unroll 4
      for (int ks = 0; ks < 32; ++ks) {
        v2f a = frag_a(A, 128, ks * 4, lane);
        v2f b = frag_b(W1, 128, ks * 4, n0, lane);
        c = wmma4(a, b, c);
      }
      for (int r = 0; r < 8; ++r) {
        int m = crow(r, lane), n = lane & 15;
        hS[wave][m * LDH + n0 + n] = silu(c[r]);
      }
    }
  }
  __syncthreads();
  if (active) {
    for (int nt = 0; nt < 24; ++nt) {
      int n0 = nt * 16;
      float bn = b2[n0 + (lane & 15)];
      v8f c;
      for (int r = 0; r < 8; ++r) c[r] = bn;
#pragma unroll 4
      for (int ks = 0; ks < 32; ++ks) {
        v2f a = frag_a(&hS[wave][0], LDH, ks * 4, lane);
        v2f b = frag_b(W2, 384, ks * 4, n0, lane);
        c = wmma4(a, b, c);
      }
      for (int r = 0; r < 8; ++r) {
        int m = crow(r, lane), n = lane & 15;
        x[(size_t)(nb + m) * 384 + n0 + n] = c[r];
      }
    }
  }
}

// ---------------- K4: edge message + scatter-add ----------------
__global__ __launch_bounds__(128) void k_edge_msg(
    const int* __restrict__ eidx, const float* __restrict__ x,
    const float* __restrict__ Wij, const float* __restrict__ dir,
    const float* __restrict__ muP, float* __restrict__ q,
    float* __restrict__ mu) {
  int e = blockIdx.x, d = threadIdx.x;
  int i = eidx[e], j = eidx[Ee + e];
  size_t xb = (size_t)j * 384, wb = (size_t)e * 384;
  float dq = x[xb + d] * Wij[wb + d];
  float dmuR = x[xb + 128 + d] * Wij[wb + 128 + d];
  float dmumu = x[xb + 256 + d] * Wij[wb + 256 + d];
  atomicAddF32(&q[(size_t)i * 128 + d], dq);
  float d0 = dir[(size_t)e * 3 + 0];
  float d1 = dir[(size_t)e * 3 + 1];
  float d2 = dir[(size_t)e * 3 + 2];
  size_t mj = (size_t)j * 384 + d, mi = (size_t)i * 384 + d;
  atomicAddF32(&mu[mi], dmuR * d0 + dmumu * muP[mj]);
  atomicAddF32(&mu[mi + 128], dmuR * d1 + dmumu * muP[mj + 128]);
  atomicAddF32(&mu[mi + 256], dmuR * d2 + dmumu * muP[mj + 256]);
}

// ---------------- K5a: [muV|muW] = mu(3N x 128) @ mix_mu_w(128x256) --------
__global__ __launch_bounds__(128) void k_mumix(const float* __restrict__ mu,
                                               const float* __restrict__ Wm,
                                               float* __restrict__ muV,
                                               float* __restrict__ muW) {
  int wave = threadIdx.x >> 5, lane = threadIdx.x & 31;
  int tile = blockIdx.x * 4 + wave;
  if (tile >= (3 * Nn) / 16) return;
  int rb = tile * 16;
  const float* A = mu + (size_t)rb * 128;
  for (int nt = 0; nt < 16; ++nt) {
    int n0 = nt * 16;
    v8f c;
    for (int r = 0; r < 8; ++r) c[r] = 0.0f;
#pragma unroll 4
    for (int ks = 0; ks < 32; ++ks) {
      v2f a = frag_a(A, 128, ks * 4, lane);
      v2f b = frag_b(Wm, 256, ks * 4, n0, lane);
      c = wmma4(a, b, c);
    }
    float* Out = (nt < 8) ? muV : muW;
    int nn0 = (nt < 8) ? n0 : (n0 - 128);
    for (int r = 0; r < 8; ++r) {
      int m = crow(r, lane), n = lane & 15;
      Out[(size_t)(rb + m) * 128 + nn0 + n] = c[r];
    }
  }
}

// ---------------- K5b: mix update (2 waves / block, 16 nodes / wave) -------
__global__ __launch_bounds__(64) void k_mix(
    const float* __restrict__ muV, const float* __restrict__ muW,
    const float* __restrict__ W1, const float* __restrict__ b1,
    const float* __restrict__ W2, const float* __restrict__ b2,
    float* __restrict__ q, float* __restrict__ mu) {
  constexpr int LDC = 260, LDH = 132;
  __shared__ float ctxS[2][16 * LDC];  // [q | mu_Vn]
  __shared__ float hS[2][16 * LDH];
  int wave = threadIdx.x >> 5, lane = threadIdx.x & 31;
  int tile = blockIdx.x * 2 + wave;  // 1250 tiles exactly
  int nb = tile * 16;
  // stage 0: build ctx = [q, ||mu_V|| ]
  for (int idx = lane; idx < 2048; idx += 32) {
    int m = idx >> 7, d = idx & 127;
    size_t mb = (size_t)(nb + m) * 384 + d;
    float sv = 0.0f;
    for (int s = 0; s < 3; ++s) {
      float v = muV[mb + s * 128];
      sv += v * v;
    }
    ctxS[wave][m * LDC + d] = q[(size_t)(nb + m) * 128 + d];
    ctxS[wave][m * LDC + 128 + d] = sqrtf(sv + 1e-8f);
  }
  __syncthreads();
  // stage 1: h = silu(ctx @ W1 + b1), K=256
  for (int nt = 0; nt < 8; ++nt) {
    int n0 = nt * 16;
    float bn = b1[n0 + (lane & 15)];
    v8f c;
    for (int r = 0; r < 8; ++r) c[r] = bn;
#pragma unroll 4
    for (int ks = 0; ks < 64; ++ks) {
      v2f a = frag_a(&ctxS[wave][0], LDC, ks * 4, lane);
      v2f b = frag_b(W1, 128, ks * 4, n0, lane);
      c = wmma4(a, b, c);
    }
    for (int r = 0; r < 8; ++r) {
      int m = crow(r, lane), n = lane & 15;
      hS[wave][m * LDH + n0 + n] = silu(c[r]);
    }
  }
  __syncthreads();
  // stage 2: xm = h @ W2 + b2 (K=128, N=384), apply updates in epilogue
  for (int nt = 0; nt < 24; ++nt) {
    int n0 = nt * 16;
    float bn = b2[n0 + (lane & 15)];
    v8f c;
    for (int r = 0; r < 8; ++r) c[r] = bn;
#pragma unroll 4
    for (int ks = 0; ks < 32; ++ks) {
      v2f a = frag_a(&hS[wave][0], LDH, ks * 4, lane);
      v2f b = frag_b(W2, 384, ks * 4, n0, lane);
      c = wmma4(a, b, c);
    }
    for (int r = 0; r < 8; ++r) {
      int m = crow(r, lane), n = lane & 15;
      int nn = n0 + n;
      size_t node = (size_t)(nb + m);
      if (nt < 8) {  // dq_i
        q[node * 128 + nn] += c[r];
      } else if (nt < 16) {  // dmu_i * mu_W
        int d = nn - 128;
        size_t mb = node * 384 + d;
        for (int s = 0; s < 3; ++s) mu[mb + s * 128] += c[r] * muW[mb + s * 128];
      } else {  // dqmu_i * sum_s(mu_V*mu_W)
        int d = nn - 256;
        size_t mb = node * 384 + d;
        float dot = 0.0f;
        for (int s = 0; s < 3; ++s) dot += muV[mb + s * 128] * muW[mb + s * 128];
        q[node * 128 + d] += c[r] * dot;
      }
    }
  }
}

// ---------------- K6: layernorm + tanh -> node_attr ----------------
__global__ __launch_bounds__(128) void k_lnorm(const float* __restrict__ q,
                                               const float* __restrict__ g,
                                               const float* __restrict__ bta,
                                               float* __restrict__ out) {
  __shared__ float red[128];
  int n = blockIdx.x, d = threadIdx.x;
  float v = q[(size_t)n * 128 + d];
  v = (v != v) ? 0.0f : fminf(fmaxf(v, -5.0f), 5.0f);
  red[d] = v;
  __syncthreads();
  for (int s = 64; s > 0; s >>= 1) {
    if (d < s) red[d] += red[d + s];
    __syncthreads();
  }
  float mean = red[0] * (1.0f / 128.0f);
  __syncthreads();
  float cdev = v - mean;
  red[d] = cdev * cdev;
  __syncthreads();
  for (int s = 64; s > 0; s >>= 1) {
    if (d < s) red[d] += red[d + s];
    __syncthreads();
  }
  float var = red[0] * (1.0f / 128.0f);
  float y = tanhf(cdev * rsqrtf(var + 1e-5f) * g[d] + bta[d]);
  out[(size_t)n * 128 + d] = (y != y) ? 0.0f : y;
}

// ---------------- K7: mu_node = segment_sum(nan0(mu)[idx_j]) ----------------
__global__ __launch_bounds__(128) void k_scatter_mu(
    const int* __restrict__ eidx, const float* __restrict__ mu,
    float* __restrict__ muN) {
  int e = blockIdx.x, d = threadIdx.x;
  int i = eidx[e], j = eidx[Ee + e];
  size_t mj = (size_t)j * 384 + d, mi = (size_t)i * 384 + d;
  for (int s = 0; s < 3; ++s) {
    float v = mu[mj + s * 128];
    v = (v != v) ? 0.0f : fminf(fmaxf(v, -3.4e38f), 3.4e38f);
    atomicAddF32(&muN[mi + s * 128], v);
  }
}

// ---------------- K8: pos_attr = clip(clip(mu_node) @ mu_proj_w) ------------
__global__ __launch_bounds__(128) void k_pos(const float* __restrict__ muN,
                                             const float* __restrict__ w,
                                             float* __restrict__ out) {
  __shared__ float red[128];
  int n = blockIdx.x, d = threadIdx.x;
  for (int s = 0; s < 3; ++s) {
    float v = muN[(size_t)n * 384 + s * 128 + d];
    v = (v != v) ? 0.0f : fminf(fmaxf(v, -5.0f), 5.0f);
    red[d] = v * w[d];
    __syncthreads();
    for (int t = 64; t > 0; t >>= 1) {
      if (d < t) red[d] += red[d + t];
      __syncthreads();
    }
    if (d == 0) {
      float p = red[0];
      p = (p != p) ? 0.0f : fminf(fmaxf(p, -5.0f), 5.0f);
      out[(size_t)n * 3 + s] = p;
    }
    __syncthreads();
  }
}

// ---------------- host launcher ----------------
extern "C" void kernel_launch(void* const* d_in, const int* in_sizes, int n_in,
                              void* d_out, int out_size, void* d_ws,
                              size_t ws_size, hipStream_t stream) {
  const float* z = (const float*)d_in[0];
  const float* pos = (const float*)d_in[1];
  const int* eidx = (const int*)d_in[2];
  const float* in_w = (const float*)d_in[3];
  const float* in_b = (const float*)d_in[4];
  const float* filt_w = (const float*)d_in[5];
  const float* filt_b = (const float*)d_in[6];
  const float* inter_w1 = (const float*)d_in[7];
  const float* inter_b1 = (const float*)d_in[8];
  const float* inter_w2 = (const float*)d_in[9];
  const float* inter_b2 = (const float*)d_in[10];
  const float* mix_mu_w = (const float*)d_in[11];
  const float* mix_w1 = (const float*)d_in[12];
  const float* mix_b1 = (const float*)d_in[13];
  const float* mix_w2 = (const float*)d_in[14];
  const float* mix_b2 = (const float*)d_in[15];
  const float* mu_proj_w = (const float*)d_in[16];
  const float* ln_g = (const float*)d_in[17];
  const float* ln_b = (const float*)d_in[18];

  float* ws = (float*)d_ws;
  size_t off = 0;
  float* Wij = ws + off;    off += (size_t)Ee * 384;  // 491.5 MB
  float* dir = ws + off;    off += (size_t)Ee * 3;
  float* q = ws + off;      off += (size_t)Nn * 128;
  float* mu = ws + off;     off += (size_t)Nn * 384;
  float* muPrev = ws + off; off += (size_t)Nn * 384;
  float* x = ws + off;      off += (size_t)Nn * 384;
  float* muV = ws + off;    off += (size_t)Nn * 384;
  float* muW = ws + off;    off += (size_t)Nn * 384;
  float* muNode = ws + off; off += (size_t)Nn * 384;  // total ~690 MB

  hipMemsetAsync(mu, 0, (size_t)Nn * 384 * sizeof(float), stream);
  hipMemsetAsync(muNode, 0, (size_t)Nn * 384 * sizeof(float), stream);

  k_edge_filters<<<Ee / 64, 128, 0, stream>>>(pos, eidx, filt_w, filt_b, Wij,
                                              dir);
  k_embed<<<(Nn / 16 + 3) / 4, 128, 0, stream>>>(z, in_w, in_b, q);

  for (int l = 0; l < Ll; ++l) {
    k_inter<<<(Nn / 16 + 3) / 4, 128, 0, stream>>>(
        q, inter_w1 + (size_t)l * 128 * 128, inter_b1 + (size_t)l * 128,
        inter_w2 + (size_t)l * 128 * 384, inter_b2 + (size_t)l * 384, x);
    hipMemcpyAsync(muPrev, mu, (size_t)Nn * 384 * sizeof(float),
                   hipMemcpyDeviceToDevice, stream);
    k_edge_msg<<<Ee, 128, 0, stream>>>(eidx, x, Wij, dir, muPrev, q, mu);
    k_mumix<<<(3 * Nn / 16 + 3) / 4, 128, 0, stream>>>(
        mu, mix_mu_w + (size_t)l * 128 * 256, muV, muW);
    k_mix<<<Nn / 32, 64, 0, stream>>>(
        muV, muW, mix_w1 + (size_t)l * 256 * 128, mix_b1 + (size_t)l * 128,
        mix_w2 + (size_t)l * 128 * 384, mix_b2 + (size_t)l * 384, q, mu);
  }

  float* out = (float*)d_out;
  k_lnorm<<<Nn, 128, 0, stream>>>(q, ln_g, ln_b, out);
  k_scatter_mu<<<Ee, 128, 0, stream>>>(eidx, mu, muNode);
  k_pos<<<Nn, 128, 0, stream>>>(muNode, mu_proj_w, out + (size_t)Nn * 128);
}